// SparseMoE_44736379355520
// MI455X (gfx1250) — compile-verified
//
#include <hip/hip_runtime.h>
#include <hip/hip_bf16.h>

typedef __attribute__((ext_vector_type(2))) float v2f;
typedef __attribute__((ext_vector_type(8))) float v8f;

#define D_EMB 768
#define F_HID 3072
#define NEXP  8

// Load one 16x4 (or 4x16) f32 WMMA fragment: uniform base + 32-bit element
// offset -> saddr-form global_load_b64.
__device__ __forceinline__ v2f frag_ld(const float* __restrict__ base, int off) {
    return *(const v2f*)(base + off);
}

// ---------------------------------------------------------------------------
// NT GEMM with f32 WMMA:  C[M,N] = act(A[M,K] @ B[N,K]^T + bias[N])
// A, B row-major, K contiguous. Workgroup: 256 threads = 8 waves (2M x 4N),
// macro-tile 128x256. Wave tile 64x64 = 4x4 WMMA tiles: 16 WMMA / 8 loads per
// k-step, software-pipelined one step ahead.
// Requires: M % 128 == 0, N % 256 == 0, K % 4 == 0.
// ---------------------------------------------------------------------------
template <bool RELU>
__global__ __launch_bounds__(256)
void gemm_nt_bias(const float* __restrict__ A, const float* __restrict__ B,
                  const float* __restrict__ bias, float* __restrict__ C,
                  int M, int N, int K)
{
    const int lane = threadIdx.x & 31;
    const int wave = threadIdx.x >> 5;
    const int wm   = wave & 1;        // 0..1
    const int wn   = wave >> 1;       // 0..3
    const int half = lane >> 4;       // K sub-slot: lanes 16-31 carry k+2,k+3
    const int r16  = lane & 15;

    const int m_base = blockIdx.y * 128 + wm * 64;
    const int n_base = blockIdx.x * 256 + wn * 64;

    // 32-bit per-lane element offsets (A: 16x4 MxK frag, B: 4x16 KxN frag with
    // B[k][n] = W[n][k] -> identical addressing).
    int offA[4], offB[4];
    #pragma unroll
    for (int i = 0; i < 4; ++i) {
        offA[i] = (m_base + i * 16 + r16) * K + 2 * half;
        offB[i] = (n_base + i * 16 + r16) * K + 2 * half;
    }

    v8f acc[4][4] = {};
    v2f ca[4], cb[4], na[4], nb[4];

    #pragma unroll
    for (int i = 0; i < 4; ++i) {
        ca[i] = frag_ld(A, offA[i]);
        cb[i] = frag_ld(B, offB[i]);
    }

    int k = 0;
    for (; k + 4 < K; k += 4) {
        // Prefetch next k-step fragments before this step's WMMAs.
        #pragma unroll
        for (int i = 0; i < 4; ++i) {
            na[i] = frag_ld(A, offA[i] + k + 4);
            nb[i] = frag_ld(B, offB[i] + k + 4);
        }
        #pragma unroll
        for (int mb = 0; mb < 4; ++mb)
            #pragma unroll
            for (int nbk = 0; nbk < 4; ++nbk)
                acc[mb][nbk] = __builtin_amdgcn_wmma_f32_16x16x4_f32(
                    false, ca[mb], false, cb[nbk], (short)0, acc[mb][nbk], false, false);
        #pragma unroll
        for (int i = 0; i < 4; ++i) { ca[i] = na[i]; cb[i] = nb[i]; }
    }
    #pragma unroll
    for (int mb = 0; mb < 4; ++mb)
        #pragma unroll
        for (int nbk = 0; nbk < 4; ++nbk)
            acc[mb][nbk] = __builtin_amdgcn_wmma_f32_16x16x4_f32(
                false, ca[mb], false, cb[nbk], (short)0, acc[mb][nbk], false, false);

    // Epilogue. C/D layout: reg r -> (m = tile_m + r + 8*half, n = tile_n + (lane&15)).
    #pragma unroll
    for (int nbk = 0; nbk < 4; ++nbk) {
        const int n  = n_base + nbk * 16 + r16;
        const float bv = bias[n];
        #pragma unroll
        for (int mb = 0; mb < 4; ++mb) {
            const int mrow0 = m_base + mb * 16 + half * 8;
            #pragma unroll
            for (int r = 0; r < 8; ++r) {
                float v = acc[mb][nbk][r] + bv;
                if (RELU) v = fmaxf(v, 0.0f);
                C[(mrow0 + r) * N + n] = v;
            }
        }
    }
}

// ---------------------------------------------------------------------------
// Expert layer 2 fused with routing weights:
//   out[m,n] (+)= w[m,e] * (mid[M,K] @ W2e[N,K]^T + b2e[n])
// ---------------------------------------------------------------------------
__global__ __launch_bounds__(256)
void gemm2_moe(const float* __restrict__ Amid, const float* __restrict__ W2e,
               const float* __restrict__ b2e, const float* __restrict__ wrt,
               float* __restrict__ out, int M, int N, int K, int e, int first)
{
    const int lane = threadIdx.x & 31;
    const int wave = threadIdx.x >> 5;
    const int wm   = wave & 1;
    const int wn   = wave >> 1;
    const int half = lane >> 4;
    const int r16  = lane & 15;

    const int m_base = blockIdx.y * 128 + wm * 64;
    const int n_base = blockIdx.x * 256 + wn * 64;

    int offA[4], offB[4];
    #pragma unroll
    for (int i = 0; i < 4; ++i) {
        offA[i] = (m_base + i * 16 + r16) * K + 2 * half;
        offB[i] = (n_base + i * 16 + r16) * K + 2 * half;
    }

    v8f acc[4][4] = {};
    v2f ca[4], cb[4], na[4], nb[4];

    #pragma unroll
    for (int i = 0; i < 4; ++i) {
        ca[i] = frag_ld(Amid, offA[i]);
        cb[i] = frag_ld(W2e, offB[i]);
    }

    int k = 0;
    for (; k + 4 < K; k += 4) {
        #pragma unroll
        for (int i = 0; i < 4; ++i) {
            na[i] = frag_ld(Amid, offA[i] + k + 4);
            nb[i] = frag_ld(W2e, offB[i] + k + 4);
        }
        #pragma unroll
        for (int mb = 0; mb < 4; ++mb)
            #pragma unroll
            for (int nbk = 0; nbk < 4; ++nbk)
                acc[mb][nbk] = __builtin_amdgcn_wmma_f32_16x16x4_f32(
                    false, ca[mb], false, cb[nbk], (short)0, acc[mb][nbk], false, false);
        #pragma unroll
        for (int i = 0; i < 4; ++i) { ca[i] = na[i]; cb[i] = nb[i]; }
    }
    #pragma unroll
    for (int mb = 0; mb < 4; ++mb)
        #pragma unroll
        for (int nbk = 0; nbk < 4; ++nbk)
            acc[mb][nbk] = __builtin_amdgcn_wmma_f32_16x16x4_f32(
                false, ca[mb], false, cb[nbk], (short)0, acc[mb][nbk], false, false);

    #pragma unroll
    for (int nbk = 0; nbk < 4; ++nbk) {
        const int n  = n_base + nbk * 16 + r16;
        const float bv = b2e[n];
        #pragma unroll
        for (int mb = 0; mb < 4; ++mb) {
            const int mrow0 = m_base + mb * 16 + half * 8;
            #pragma unroll
            for (int r = 0; r < 8; ++r) {
                const int m = mrow0 + r;
                const float y  = acc[mb][nbk][r] + bv;
                const float wt = wrt[m * NEXP + e];
                const int  o   = m * N + n;
                if (first) out[o] = wt * y;
                else       out[o] += wt * y;
            }
        }
    }
}

// ---------------------------------------------------------------------------
// Router tail: score = h @ Wr2^T + br2 ; top-2 ; masked softmax -> w[T,8].
// One wave32 per token.
// ---------------------------------------------------------------------------
__global__ __launch_bounds__(256)
void router_topk(const float* __restrict__ h, const float* __restrict__ Wr2,
                 const float* __restrict__ br2, float* __restrict__ w, int T)
{
    const int gwave = (int)((blockIdx.x * blockDim.x + threadIdx.x) >> 5);
    const int lane  = threadIdx.x & 31;
    if (gwave >= T) return;

    const float* ht = h + (size_t)gwave * D_EMB;
    float s[NEXP] = {0.f, 0.f, 0.f, 0.f, 0.f, 0.f, 0.f, 0.f};

    for (int d = lane; d < D_EMB; d += 32) {
        const float hv = ht[d];
        #pragma unroll
        for (int e = 0; e < NEXP; ++e)
            s[e] += hv * Wr2[e * D_EMB + d];
    }

    #pragma unroll
    for (int e = 0; e < NEXP; ++e) {
        #pragma unroll
        for (int off = 16; off > 0; off >>= 1)
            s[e] += __shfl_xor(s[e], off, 32);
        s[e] += br2[e];
    }

    // top-2 (strict > keeps lowest index on ties, matching lax.top_k)
    int i1 = 0; float s1 = s[0];
    #pragma unroll
    for (int e = 1; e < NEXP; ++e)
        if (s[e] > s1) { s1 = s[e]; i1 = e; }
    int i2 = -1; float s2 = -3.402823e38f;
    #pragma unroll
    for (int e = 0; e < NEXP; ++e)
        if (e != i1 && s[e] > s2) { s2 = s[e]; i2 = e; }

    const float e2  = __expf(s2 - s1);
    const float inv = 1.0f / (1.0f + e2);

    if (lane < NEXP) {
        float wv = 0.0f;
        if (lane == i1) wv = inv;
        else if (lane == i2) wv = e2 * inv;
        w[gwave * NEXP + lane] = wv;
    }
}

// ---------------------------------------------------------------------------
extern "C" void kernel_launch(void* const* d_in, const int* in_sizes, int n_in,
                              void* d_out, int out_size, void* d_ws, size_t ws_size,
                              hipStream_t stream)
{
    const float* x   = (const float*)d_in[0];
    const float* W1  = (const float*)d_in[1];
    const float* b1  = (const float*)d_in[2];
    const float* W2  = (const float*)d_in[3];
    const float* b2  = (const float*)d_in[4];
    const float* Wr1 = (const float*)d_in[5];
    const float* br1 = (const float*)d_in[6];
    const float* Wr2 = (const float*)d_in[7];
    const float* br2 = (const float*)d_in[8];
    float* out = (float*)d_out;

    const int T = in_sizes[0] / D_EMB;   // 4096 tokens, multiple of 128

    // Workspace layout: h[T,D] | w[T,8] | mid[Mc,F]
    float* h   = (float*)d_ws;
    float* w   = h + (size_t)T * D_EMB;
    float* mid = w + (size_t)T * NEXP;
    const size_t used  = ((size_t)T * D_EMB + (size_t)T * NEXP) * sizeof(float);
    const size_t avail = (ws_size > used) ? (ws_size - used) : 0;
    int Mc = (int)(avail / ((size_t)F_HID * sizeof(float)));
    Mc = (Mc / 128) * 128;
    if (Mc > T)   Mc = T;
    if (Mc < 128) Mc = 128;

    // Router layer 1: h = relu(x @ Wr1^T + br1)   [T, D]
    gemm_nt_bias<true><<<dim3(D_EMB / 256, T / 128), 256, 0, stream>>>(
        x, Wr1, br1, h, T, D_EMB, D_EMB);

    // Router layer 2 + top-2 + masked softmax -> w [T, 8]
    router_topk<<<(T + 7) / 8, 256, 0, stream>>>(h, Wr2, br2, w, T);

    // Experts (dense-with-zero-weights == routed result), chunked over tokens.
    for (int m0 = 0; m0 < T; m0 += Mc) {
        const int Mcur = (T - m0 < Mc) ? (T - m0) : Mc;
        for (int e = 0; e < NEXP; ++e) {
            // mid = relu(x_chunk @ W1[e]^T + b1[e])   [Mcur, F]
            gemm_nt_bias<true><<<dim3(F_HID / 256, Mcur / 128), 256, 0, stream>>>(
                x + (size_t)m0 * D_EMB,
                W1 + (size_t)e * F_HID * D_EMB,
                b1 + (size_t)e * F_HID,
                mid, Mcur, F_HID, D_EMB);
            // out_chunk (+)= w[:,e] * (mid @ W2[e]^T + b2[e])
            gemm2_moe<<<dim3(D_EMB / 256, Mcur / 128), 256, 0, stream>>>(
                mid,
                W2 + (size_t)e * D_EMB * F_HID,
                b2 + (size_t)e * D_EMB,
                w + (size_t)m0 * NEXP,
                out + (size_t)m0 * D_EMB,
                Mcur, D_EMB, F_HID, e, (e == 0) ? 1 : 0);
        }
    }
}